// InteractionNetwork_39333310497017
// MI455X (gfx1250) — compile-verified
//
#include <hip/hip_runtime.h>
#include <hip/hip_bf16.h>

// ---------------------------------------------------------------------------
// fp32 WMMA types (CDNA5 / gfx1250, wave32)
// A (16x4 f32) and B (4x16 f32): 2 VGPRs/lane.  C/D (16x16 f32): 8 VGPRs/lane.
// ---------------------------------------------------------------------------
typedef __attribute__((ext_vector_type(2))) float v2f;
typedef __attribute__((ext_vector_type(4))) float v4f;
typedef __attribute__((ext_vector_type(8))) float v8f;
typedef __attribute__((ext_vector_type(4))) int   v4i;

// ---------------------------------------------------------------------------
// Async global -> LDS copy (16 bytes per lane), with sync fallback.
// Builtin signature (from hipcc diagnostic): arg0 = v4i* in addrspace(1)
// (global source), arg1 = v4i* in addrspace(3) (LDS dest), then imm offset
// and imm cpol.
// ---------------------------------------------------------------------------
#if defined(__has_builtin)
#if __has_builtin(__builtin_amdgcn_global_load_async_to_lds_b128)
#define USE_ASYNC_LDS 1
#endif
#if __has_builtin(__builtin_amdgcn_s_wait_asynccnt)
#define HAVE_WAIT_ASYNC_BUILTIN 1
#endif
#endif

__device__ __forceinline__ void lds_copy16(const float* g, float* l)
{
#if defined(USE_ASYNC_LDS)
    __builtin_amdgcn_global_load_async_to_lds_b128(
        (__attribute__((address_space(1))) v4i*)g,
        (__attribute__((address_space(3))) v4i*)l,
        /*imm_offset=*/0, /*cpol=*/0);
#else
    *(v4f*)l = *(const v4f*)g;
#endif
}

__device__ __forceinline__ void wait_async_copies()
{
#if defined(USE_ASYNC_LDS)
#if defined(HAVE_WAIT_ASYNC_BUILTIN)
    __builtin_amdgcn_s_wait_asynccnt(0);
#else
    asm volatile("s_wait_asynccnt 0x0" ::: "memory");
#endif
#endif
}

// ---------------------------------------------------------------------------
// Generic batched GEMM: C[b] = act(A[b] @ B[b] + bias)
//   A: row-major [M,K], lda
//   B: BT==false -> row-major [K,N], ldb ; BT==true -> B[k][n] = Bt[n*ldb + k]
//   C: row-major [M,N], ldc
//
// Block = 256 threads = 8 waves arranged 4(M) x 2(N); wave tile = 16x64.
// Workgroup tile = 64(M) x 128(N), K-chunk = 16, double-buffered in LDS with
// async global->LDS copies.  Requires: M%64==0, N%128==0, K%16==0, and all
// row pointers 16B-aligned (lda/ldb multiples of 4).
// ---------------------------------------------------------------------------
template <bool BT, bool RELU, bool BIAS>
__global__ __launch_bounds__(256) void gemm_wmma_f32(
    const float* __restrict__ A, int lda, long long strideA,
    const float* __restrict__ B, int ldb, long long strideB,
    const float* __restrict__ bias,
    float* __restrict__ C, int ldc, long long strideC,
    int K)
{
    // Padded LDS tiles.  Row strides chosen so 16B async writes stay aligned
    // (20*4=80B, 132*4=528B, both multiples of 16) and fragment reads are
    // bank-conflict-free (stride 20 mod 64 spans {0,4,...,60}).
    constexpr int ATILE = 64 * 20;
    constexpr int BTILE = BT ? 128 * 20 : 16 * 132;

    __shared__ float As[2][ATILE];
    __shared__ float Bs[2][BTILE];

    const int tid  = threadIdx.x;
    const int lane = tid & 31;
    const int wave = tid >> 5;
    const int waveM = wave & 3;   // 0..3
    const int waveN = wave >> 2;  // 0..1

    const int m0b = blockIdx.x * 64;
    const int n0b = blockIdx.y * 128;

    const long long bz = blockIdx.z;
    A += bz * strideA;
    B += bz * strideB;
    C += bz * strideC;

    const int kh  = lane >> 4;   // 0: K pair {0,1}; 1: K pair {2,3}
    const int l16 = lane & 15;

    v8f acc[4];
    acc[0] = v8f{}; acc[1] = v8f{}; acc[2] = v8f{}; acc[3] = v8f{};

    // --- tile copy helpers (256 threads cooperate, 16B per transfer) -------
    auto copyA = [&](int k0, int buf) {
        // 64 rows x 16 k = 256 x 16B
        const int row = tid >> 2;
        const int seg = (tid & 3) * 4;
        lds_copy16(A + (m0b + row) * lda + k0 + seg, &As[buf][row * 20 + seg]);
    };
    auto copyB = [&](int k0, int buf) {
#pragma unroll
        for (int i = 0; i < 2; ++i) {
            const int q = tid + i * 256;   // 512 x 16B
            if (BT) {
                // Bt rows are k-contiguous: tile stored [n][k] (stride 20)
                const int col = q >> 2;
                const int seg = (q & 3) * 4;
                lds_copy16(B + (n0b + col) * ldb + k0 + seg,
                           &Bs[buf][col * 20 + seg]);
            } else {
                // row-major B: tile stored [k][n] (stride 132)
                const int row = q >> 5;
                const int seg = (q & 31) * 4;
                lds_copy16(B + (k0 + row) * ldb + n0b + seg,
                           &Bs[buf][row * 132 + seg]);
            }
        }
    };

    const int nch = K >> 4;        // K / 16
    copyA(0, 0);
    copyB(0, 0);

    const int rowL = waveM * 16 + l16;   // local A row for this lane
    const int nL   = waveN * 64 + l16;   // local B col base for this lane

    for (int c = 0; c < nch; ++c) {
        const int cur = c & 1;

        wait_async_copies();   // chunk c resident (each wave waits its own)
        __syncthreads();       // ... and visible to all waves

        if (c + 1 < nch) {     // prefetch chunk c+1 into the other buffer
            copyA((c + 1) << 4, cur ^ 1);
            copyB((c + 1) << 4, cur ^ 1);
        }

        const float* as = As[cur];
        const float* bs = Bs[cur];

#pragma unroll
        for (int kk = 0; kk < 16; kk += 4) {
            const int ka = kk + kh * 2;

            v2f a;
            a.x = as[rowL * 20 + ka];
            a.y = as[rowL * 20 + ka + 1];

            v2f b[4];
#pragma unroll
            for (int t = 0; t < 4; ++t) {
                if (BT) {
                    b[t].x = bs[(nL + t * 16) * 20 + ka];
                    b[t].y = bs[(nL + t * 16) * 20 + ka + 1];
                } else {
                    b[t].x = bs[ka * 132 + nL + t * 16];
                    b[t].y = bs[(ka + 1) * 132 + nL + t * 16];
                }
            }

#pragma unroll
            for (int t = 0; t < 4; ++t) {
                acc[t] = __builtin_amdgcn_wmma_f32_16x16x4_f32(
                    /*neg_a=*/false, a, /*neg_b=*/false, b[t],
                    /*c_mod=*/(short)0, acc[t],
                    /*reuse_a=*/false, /*reuse_b=*/false);
            }
        }
    }

    // Epilogue: C/D layout -> VGPR j holds row (m0 + j + 8*kh), col n0+t*16+l16
    const int mBase = m0b + waveM * 16 + kh * 8;
    const int n0w   = n0b + waveN * 64;
#pragma unroll
    for (int t = 0; t < 4; ++t) {
        const int n = n0w + t * 16 + l16;
        float bs = 0.0f;
        if (BIAS) bs = bias[n];
#pragma unroll
        for (int j = 0; j < 8; ++j) {
            float v = acc[t][j] + bs;
            if (RELU) v = fmaxf(v, 0.0f);
            C[(mBase + j) * ldc + n] = v;
        }
    }
}

// ---------------------------------------------------------------------------
// Copy ra [8,64,8192] into relation_input rows 256:320 (flat contiguous per b)
// ---------------------------------------------------------------------------
__global__ __launch_bounds__(256) void copy_ra_kernel(
    const float* __restrict__ ra, float* __restrict__ RI)
{
    const long long i = (long long)blockIdx.x * 256 + threadIdx.x; // < 8*64*8192
    const long long per_b = 64LL * 8192;
    const long long b = i / per_b;
    const long long off = i - b * per_b;
    RI[b * (320LL * 8192) + 256LL * 8192 + off] = ra[i];
}

// ---------------------------------------------------------------------------
// aggregated[b, 0:400, n] = concat(objects[b,0:128,n], externals[b,0:16,n],
//                                  agg_eff[b,0:256,n])
// ---------------------------------------------------------------------------
__global__ __launch_bounds__(256) void build_agg_kernel(
    const float* __restrict__ objects,   // [8,128,1024]
    const float* __restrict__ externals, // [8,16,1024]
    const float* __restrict__ agge,      // [8,256,1024]
    float* __restrict__ out)             // [8,400,1024]
{
    const long long i = (long long)blockIdx.x * 256 + threadIdx.x; // < 8*400*1024
    const int per_b = 400 * 1024;
    const int b = (int)(i / per_b);
    const int rem = (int)(i - (long long)b * per_b);
    const int c = rem >> 10;      // / 1024
    const int n = rem & 1023;     // % 1024
    float v;
    if (c < 128)       v = objects[(b * 128 + c) * 1024 + n];
    else if (c < 144)  v = externals[(b * 16 + (c - 128)) * 1024 + n];
    else               v = agge[(b * 256 + (c - 144)) * 1024 + n];
    out[i] = v;
}

// ---------------------------------------------------------------------------
// Row sum: prediction flat viewed as [1024 rows][1024 cols] (row = b*128+d),
// sums[row] = sum_n pred[row*1024 + n]
// ---------------------------------------------------------------------------
__global__ __launch_bounds__(256) void rowsum_kernel(
    const float* __restrict__ pred, float* __restrict__ sums)
{
    __shared__ float sdata[256];
    const int row = blockIdx.x;
    const int tid = threadIdx.x;
    float s = 0.0f;
    for (int n = tid; n < 1024; n += 256)
        s += pred[row * 1024 + n];
    sdata[tid] = s;
    __syncthreads();
    for (int stride = 128; stride > 0; stride >>= 1) {
        if (tid < stride) sdata[tid] += sdata[tid + stride];
        __syncthreads();
    }
    if (tid == 0) sums[row] = sdata[0];
}

// ---------------------------------------------------------------------------
// Tiny attribute MLP: [8,128] -> relu(@W1[128,64]+b1) -> relu(@W2[64,32]+b2)
// ---------------------------------------------------------------------------
__global__ __launch_bounds__(256) void attr_mlp_kernel(
    const float* __restrict__ sums, // [8,128]
    const float* __restrict__ W1, const float* __restrict__ b1,
    const float* __restrict__ W2, const float* __restrict__ b2,
    float* __restrict__ out)        // [8,32]
{
    __shared__ float h0[8 * 128];
    __shared__ float h1[8 * 64];
    const int tid = threadIdx.x;
    for (int i = tid; i < 8 * 128; i += 256) h0[i] = sums[i];
    __syncthreads();
    for (int o = tid; o < 8 * 64; o += 256) {
        const int b = o >> 6, c = o & 63;
        float acc = b1[c];
        for (int k = 0; k < 128; ++k) acc += h0[b * 128 + k] * W1[k * 64 + c];
        h1[o] = fmaxf(acc, 0.0f);
    }
    __syncthreads();
    for (int o = tid; o < 8 * 32; o += 256) {
        const int b = o >> 5, c = o & 31;
        float acc = b2[c];
        for (int k = 0; k < 64; ++k) acc += h1[b * 64 + k] * W2[k * 32 + c];
        out[o] = fmaxf(acc, 0.0f);
    }
}

// ---------------------------------------------------------------------------
// Host-side launcher
// ---------------------------------------------------------------------------
extern "C" void kernel_launch(void* const* d_in, const int* in_sizes, int n_in,
                              void* d_out, int out_size, void* d_ws, size_t ws_size,
                              hipStream_t stream)
{
    (void)in_sizes; (void)n_in; (void)out_size; (void)ws_size;

    const float* objects   = (const float*)d_in[0];  // [8,128,1024]
    const float* externals = (const float*)d_in[1];  // [8,16,1024]
    const float* rs        = (const float*)d_in[2];  // [8,1024,8192]
    const float* rr        = (const float*)d_in[3];  // [8,1024,8192]
    const float* ra        = (const float*)d_in[4];  // [8,64,8192]
    const float* relW1 = (const float*)d_in[5],  *relb1 = (const float*)d_in[6];
    const float* relW2 = (const float*)d_in[7],  *relb2 = (const float*)d_in[8];
    const float* relW3 = (const float*)d_in[9],  *relb3 = (const float*)d_in[10];
    const float* objW1 = (const float*)d_in[11], *objb1 = (const float*)d_in[12];
    const float* objW2 = (const float*)d_in[13], *objb2 = (const float*)d_in[14];
    const float* objW3 = (const float*)d_in[15], *objb3 = (const float*)d_in[16];
    const float* atW1  = (const float*)d_in[17], *atb1  = (const float*)d_in[18];
    const float* atW2  = (const float*)d_in[19], *atb2  = (const float*)d_in[20];

    float* ws = (float*)d_ws;
    float* region0 = ws;                       // 33,554,432 floats (134 MB)
    float* region1 = ws + 33554432LL;          // 33,554,432 floats (134 MB)

    float* RI   = region0;                 // relation_input flat [65536,320]
    float* H1   = region1;                 // [65536,512]
    float* H2   = region0;                 // [65536,512]  (RI dead)
    float* EFF  = region1;                 // [65536,256] == [8,256,8192] (H1 dead)
    float* AGGE = region0;                 // [8,256,1024] (H2 dead)
    float* AGGR = region0 + 4194304LL;     // [8,400,1024]
    float* OH1  = region1;                 // [8192,512]   (EFF dead)
    float* OH2  = region1 + 8388608LL;     // [8192,256]
    float* SUMS = region0;                 // [8,128]      (AGGR dead)

    float* pred = (float*)d_out;           // [8,128,1024] flat (== [8192,128] rows)
    float* attr = (float*)d_out + 1048576; // [8,32]

    const dim3 blk(256);

    // 1) ra -> relation_input rows 256:320
    copy_ra_kernel<<<16384, blk, 0, stream>>>(ra, RI);

    // 2) sender = objects @ rs   -> RI rows 0:128   (per-b GEMM 128x8192x1024)
    gemm_wmma_f32<false, false, false><<<dim3(2, 64, 8), blk, 0, stream>>>(
        objects, 1024, 131072LL, rs, 8192, 8388608LL, nullptr,
        RI, 8192, 2621440LL, 1024);

    // 3) receiver = objects @ rr -> RI rows 128:256
    gemm_wmma_f32<false, false, false><<<dim3(2, 64, 8), blk, 0, stream>>>(
        objects, 1024, 131072LL, rr, 8192, 8388608LL, nullptr,
        RI + 128LL * 8192, 8192, 2621440LL, 1024);

    // 4) rel MLP (flat reinterpret rows of 320): [65536,320]->512->512->256
    gemm_wmma_f32<false, true, true><<<dim3(1024, 4, 1), blk, 0, stream>>>(
        RI, 320, 0, relW1, 512, 0, relb1, H1, 512, 0, 320);
    gemm_wmma_f32<false, true, true><<<dim3(1024, 4, 1), blk, 0, stream>>>(
        H1, 512, 0, relW2, 512, 0, relb2, H2, 512, 0, 512);
    gemm_wmma_f32<false, true, true><<<dim3(1024, 2, 1), blk, 0, stream>>>(
        H2, 512, 0, relW3, 256, 0, relb3, EFF, 256, 0, 512);

    // 5) agg_eff[b] = effects[b] (256x8192) @ rs[b]^T (8192x1024)
    gemm_wmma_f32<true, false, false><<<dim3(4, 8, 8), blk, 0, stream>>>(
        EFF, 8192, 2097152LL, rs, 8192, 8388608LL, nullptr,
        AGGE, 1024, 262144LL, 8192);

    // 6) aggregated = concat(objects, externals, agg_eff)
    build_agg_kernel<<<12800, blk, 0, stream>>>(objects, externals, AGGE, AGGR);

    // 7) obj MLP (flat rows of 400): [8192,400]->512->256->128 (to d_out)
    gemm_wmma_f32<false, true, true><<<dim3(128, 4, 1), blk, 0, stream>>>(
        AGGR, 400, 0, objW1, 512, 0, objb1, OH1, 512, 0, 400);
    gemm_wmma_f32<false, true, true><<<dim3(128, 2, 1), blk, 0, stream>>>(
        OH1, 512, 0, objW2, 256, 0, objb2, OH2, 256, 0, 512);
    gemm_wmma_f32<false, true, true><<<dim3(128, 1, 1), blk, 0, stream>>>(
        OH2, 256, 0, objW3, 128, 0, objb3, pred, 128, 0, 256);

    // 8) prediction.sum(axis=2) -> [8,128]
    rowsum_kernel<<<1024, blk, 0, stream>>>(pred, SUMS);

    // 9) attribute MLP -> d_out tail
    attr_mlp_kernel<<<1, blk, 0, stream>>>(SUMS, atW1, atb1, atW2, atb2, attr);
}